// DepthRasterization_50689204027929
// MI455X (gfx1250) — compile-verified
//
#include <hip/hip_runtime.h>
#include <hip/hip_bf16.h>

#define H_RENDER   640
#define W_RENDER   640
#define OUT_H      128
#define OUT_W      128
#define NUM_VERTS  778
#define NUM_FACES  1538
#define BATCH      4
#define NSLICES    8
#define BG_DEPTH   1.0e10f

typedef __attribute__((ext_vector_type(2))) float v2f;
typedef __attribute__((ext_vector_type(8))) float v8f;

// ---------------------------------------------------------------------------
// Pass 1: init the 128x128x4 z-buffer (== d_out) to BG_DEPTH (uint bitcast).
// ---------------------------------------------------------------------------
__global__ void zbuf_init_kernel(unsigned int* __restrict__ zbits, int n) {
    int i = blockIdx.x * blockDim.x + threadIdx.x;
    if (i < n) zbits[i] = __float_as_uint(BG_DEPTH);
}

// ---------------------------------------------------------------------------
// Pass 2: tile rasterizer. One wave32 owns one 16x16 sample tile.
// Sample (i,j) of the output corresponds to render pixel center
// (5j+2.5, 5i+2.5)  [exact-decimation equivalence of the linear resize].
// Edge/plane evaluation over the tile via V_WMMA_F32_16X16X4_F32.
// ---------------------------------------------------------------------------
__global__ __launch_bounds__(32)
void raster_tile_kernel(const float* __restrict__ verts,
                        const long long* __restrict__ faces,
                        unsigned int* __restrict__ zbits) {
    const int lane  = threadIdx.x;          // 0..31
    const int tile  = blockIdx.x;           // 0..63  (8x8 tiles of 16x16)
    const int b     = blockIdx.y;           // batch
    const int slice = blockIdx.z;           // face slice

    const int tx0 = (tile & 7) << 4;        // tile origin in 128x128 lattice
    const int ty0 = (tile >> 3) << 4;

    const int n    = lane & 15;             // column within tile
    const int half = lane >> 4;             // accumulator half (M+8 rows)
    const float lowmask = half ? 0.0f : 1.0f;

    // Sample-lattice -> render coordinates
    const float px = 5.0f * (float)(tx0 + n) + 2.5f;   // col coordinate (K=0 row of B)

    // A matrix (16x4 f32): lanes 0..15 hold row m = lane, elements K=0 (=1) and
    // K=1 (=py(m)); lanes 16..31 hold K=2,3 which we keep zero.
    v2f amat;
    amat.x = lowmask;                                          // K=0 : 1
    amat.y = lowmask * (5.0f * (float)(ty0 + lane) + 2.5f);    // K=1 : py(m)

    // Tile bounds in render coords for bbox culling
    const float tpx0 = 5.0f * (float)tx0 + 2.5f;
    const float tpx1 = 5.0f * (float)(tx0 + 15) + 2.5f;
    const float tpy0 = 5.0f * (float)ty0 + 2.5f;
    const float tpy1 = 5.0f * (float)(ty0 + 15) + 2.5f;

    float zmin[8];
#pragma unroll
    for (int v = 0; v < 8; ++v) zmin[v] = BG_DEPTH;

    const float* vb = verts + (size_t)b * NUM_VERTS * 3;

    for (int f = slice; f < NUM_FACES; f += NSLICES) {
        // gfx1250 prefetch of the upcoming face indices
        __builtin_prefetch((const void*)(faces + 3 * (f + NSLICES)), 0, 1);

        const long long i0 = faces[3 * f + 0];
        const long long i1 = faces[3 * f + 1];
        const long long i2 = faces[3 * f + 2];
        const float* p0 = vb + 3 * (size_t)i0;
        const float* p1 = vb + 3 * (size_t)i1;
        const float* p2 = vb + 3 * (size_t)i2;
        const float x0 = p0[0], y0 = p0[1], z0 = p0[2];
        const float x1 = p1[0], y1 = p1[1], z1 = p1[2];
        const float x2 = p2[0], y2 = p2[1], z2 = p2[2];

        // Scalar bbox cull (keep EXEC all-ones around WMMA)
        const float minx = fminf(x0, fminf(x1, x2));
        const float maxx = fmaxf(x0, fmaxf(x1, x2));
        const float miny = fminf(y0, fminf(y1, y2));
        const float maxy = fmaxf(y0, fmaxf(y1, y2));
        const float area = (x1 - x0) * (y2 - y0) - (y1 - y0) * (x2 - x0);
        int skip = (minx > tpx1) | (maxx < tpx0) | (miny > tpy1) | (maxy < tpy0)
                 | (fabsf(area) <= 1e-12f);
        skip = __builtin_amdgcn_readfirstlane(skip);
        if (skip) continue;

        const float s        = (area > 0.0f) ? 1.0f : -1.0f;
        const float inv_area = 1.0f / area;

        // Edge plane coefficients: e = A_e*px + B_e*py + C_e
        const float A0 = -(y2 - y1), B0 = (x2 - x1), C0 = (y2 - y1) * x1 - (x2 - x1) * y1;
        const float A1 = -(y0 - y2), B1 = (x0 - x2), C1 = (y0 - y2) * x2 - (x0 - x2) * y2;
        const float A2 = -(y1 - y0), B2 = (x1 - x0), C2 = (y1 - y0) * x0 - (x1 - x0) * y0;

        // B matrices (4x16 f32): row K=0 = A_e*px(n)+C_e, row K=1 = B_e
        v2f bm0, bm1, bm2;
        bm0.x = lowmask * (A0 * px + C0); bm0.y = lowmask * B0;
        bm1.x = lowmask * (A1 * px + C1); bm1.y = lowmask * B1;
        bm2.x = lowmask * (A2 * px + C2); bm2.y = lowmask * B2;

        v8f cz = {};
        // D = A x B + 0 : 256 plane evaluations per v_wmma_f32_16x16x4_f32
        v8f e0 = __builtin_amdgcn_wmma_f32_16x16x4_f32(
            false, amat, false, bm0, (short)0, cz, false, false);
        v8f e1 = __builtin_amdgcn_wmma_f32_16x16x4_f32(
            false, amat, false, bm1, (short)0, cz, false, false);
        v8f e2 = __builtin_amdgcn_wmma_f32_16x16x4_f32(
            false, amat, false, bm2, (short)0, cz, false, false);

#pragma unroll
        for (int v = 0; v < 8; ++v) {
            const float E0 = e0[v], E1 = e1[v], E2 = e2[v];
            const bool inside = (E0 * s >= 0.0f) && (E1 * s >= 0.0f) && (E2 * s >= 0.0f);
            const float zp = (E0 * z0 + E1 * z1 + E2 * z2) * inv_area;
            zmin[v] = fminf(zmin[v], inside ? zp : BG_DEPTH);
        }
    }

    // Commit tile: positive-float ordering == uint ordering, min is
    // order-independent -> deterministic across replays.
#pragma unroll
    for (int v = 0; v < 8; ++v) {
        const int row = ty0 + v + (half << 3);
        const int col = tx0 + n;
        atomicMin(&zbits[((size_t)b * OUT_H + row) * OUT_W + col],
                  __float_as_uint(zmin[v]));
    }
}

// ---------------------------------------------------------------------------
// Pass 3: bitcast back to float and clamp at 100.0 (reference: min(dm, 100)).
// ---------------------------------------------------------------------------
__global__ void clamp_kernel(float* __restrict__ out, int n) {
    int i = blockIdx.x * blockDim.x + threadIdx.x;
    if (i < n) {
        float v = __uint_as_float(((unsigned int*)out)[i]);
        out[i] = fminf(v, 100.0f);
    }
}

extern "C" void kernel_launch(void* const* d_in, const int* in_sizes, int n_in,
                              void* d_out, int out_size, void* d_ws, size_t ws_size,
                              hipStream_t stream) {
    (void)in_sizes; (void)n_in; (void)d_ws; (void)ws_size;
    const float*     verts = (const float*)d_in[0];       // (4, 778, 3) f32
    const long long* faces = (const long long*)d_in[1];   // (1538, 3) i64
    unsigned int*    zbits = (unsigned int*)d_out;        // (4, 128, 128)

    const int n = BATCH * OUT_H * OUT_W;                  // 65536

    zbuf_init_kernel<<<(n + 255) / 256, 256, 0, stream>>>(zbits, n);

    dim3 grid(64, BATCH, NSLICES);                        // 2048 wave32 tiles
    raster_tile_kernel<<<grid, 32, 0, stream>>>(verts, faces, zbits);

    clamp_kernel<<<(n + 255) / 256, 256, 0, stream>>>((float*)d_out, n);
}